// AdaptiveKernelFusion_42082089566428
// MI455X (gfx1250) — compile-verified
//
#include <hip/hip_runtime.h>
#include <math.h>

typedef __attribute__((ext_vector_type(2))) float v2f;
typedef __attribute__((ext_vector_type(4))) float v4f;
typedef __attribute__((ext_vector_type(8))) float v8f;

// ---------------- workspace layout (float offsets) ----------------
// 3 modalities * 8 heads * 2048 rows = 49152 per-row partials
static constexpr size_t O_RSUM  = 0;                    // 49152
static constexpr size_t O_RSQ   = 49152;                // 49152
static constexpr size_t O_RENT  = 98304;                // 49152
static constexpr size_t O_RCNT  = 147456;               // 49152
static constexpr size_t O_ALPHA = 196608;               // 16 (alpha[3])
static constexpr size_t O_DINV  = 196640;               // 16384 (8*2048)
static constexpr size_t O_HS    = 213024;               // 262144 (2048*128)
static constexpr size_t O_PART  = 475168;               // 2097152 (8*2048*128)
// total ~= 2,572,320 floats ~= 10.3 MB

static constexpr float THRESH  = 1.0f / 33554432.0f;    // 1/(H*N*N)
static constexpr float EPS_ENT = 1e-9f;

// ================= pass 1: per-row stats over all 3 attn tensors =============
// grid = 3*8*2048 = 49152 blocks, 256 threads; one block per attention row.
__global__ void __launch_bounds__(256)
k_stats(const float* __restrict__ a0, const float* __restrict__ a1,
        const float* __restrict__ a2, float* __restrict__ ws) {
  __shared__ float red[256];
  const int b   = blockIdx.x;          // m*16384 + (h*2048 + i)
  const int m   = b >> 14;
  const int hi  = b & 16383;
  const float* A = (m == 0) ? a0 : (m == 1) ? a1 : a2;
  const float* row = A + (size_t)hi * 2048;
  const int tid = threadIdx.x;

  // 8 elements per thread as two float4s (non-temporal: 402MB stream)
  v4f v0 = __builtin_nontemporal_load(((const v4f*)row) + tid * 2 + 0);
  v4f v1 = __builtin_nontemporal_load(((const v4f*)row) + tid * 2 + 1);

  float s  = v0.x + v0.y + v0.z + v0.w + v1.x + v1.y + v1.z + v1.w;
  float sq = v0.x*v0.x + v0.y*v0.y + v0.z*v0.z + v0.w*v0.w
           + v1.x*v1.x + v1.y*v1.y + v1.z*v1.z + v1.w*v1.w;
  float c  = (v0.x < THRESH) + (v0.y < THRESH) + (v0.z < THRESH) + (v0.w < THRESH)
           + (v1.x < THRESH) + (v1.y < THRESH) + (v1.z < THRESH) + (v1.w < THRESH);

  auto blockReduce = [&](float v) -> float {
    __syncthreads();
    red[tid] = v; __syncthreads();
    for (int off = 128; off > 0; off >>= 1) {
      if (tid < off) red[tid] += red[tid + off];
      __syncthreads();
    }
    return red[0];
  };

  const float rowsum = blockReduce(s);
  const float inv = 1.0f / (rowsum + EPS_ENT);
  float e = 0.0f;
  {
    float p;
    p = v0.x * inv; e -= p * logf(p + EPS_ENT);
    p = v0.y * inv; e -= p * logf(p + EPS_ENT);
    p = v0.z * inv; e -= p * logf(p + EPS_ENT);
    p = v0.w * inv; e -= p * logf(p + EPS_ENT);
    p = v1.x * inv; e -= p * logf(p + EPS_ENT);
    p = v1.y * inv; e -= p * logf(p + EPS_ENT);
    p = v1.z * inv; e -= p * logf(p + EPS_ENT);
    p = v1.w * inv; e -= p * logf(p + EPS_ENT);
  }
  const float rowent = blockReduce(e);
  const float rowsq  = blockReduce(sq);
  const float rowcnt = blockReduce(c);

  if (tid == 0) {
    ws[O_RSUM + b] = rowsum;
    ws[O_RSQ  + b] = rowsq;
    ws[O_RENT + b] = rowent;
    ws[O_RCNT + b] = rowcnt;
  }
}

// ================= pass 2a: H_split = (e0+e1+e2)/3 =====================
__global__ void k_hsplit(const float* __restrict__ e0, const float* __restrict__ e1,
                         const float* __restrict__ e2, float* __restrict__ ws) {
  int i = blockIdx.x * blockDim.x + threadIdx.x;  // float4 index, 65536 total
  if (i < 65536) {
    v4f a = ((const v4f*)e0)[i];
    v4f b = ((const v4f*)e1)[i];
    v4f c = ((const v4f*)e2)[i];
    v4f r = (a + b + c) * (1.0f / 3.0f);
    ((v4f*)(ws + O_HS))[i] = r;
  }
}

// ================= pass 2b: stats reduce + gate MLP + softmax + dinv =========
// single block, 256 threads. Deterministic (no atomics).
__global__ void __launch_bounds__(256)
k_gate(const float* __restrict__ gw1, const float* __restrict__ gb1,
       const float* __restrict__ gw2, const float* __restrict__ gb2,
       float* __restrict__ ws) {
  __shared__ float red[256];
  __shared__ float stats_s[3][4];
  __shared__ float alpha_s[3];
  const int tid = threadIdx.x;

  auto blockReduce = [&](float v) -> float {
    __syncthreads();
    red[tid] = v; __syncthreads();
    for (int off = 128; off > 0; off >>= 1) {
      if (tid < off) red[tid] += red[tid + off];
      __syncthreads();
    }
    return red[0];
  };

  for (int m = 0; m < 3; ++m) {
    float s = 0.f, q = 0.f, e = 0.f, c = 0.f;
    const size_t base = (size_t)m * 16384;
    for (int i = tid; i < 16384; i += 256) {
      s += ws[O_RSUM + base + i];
      q += ws[O_RSQ  + base + i];
      e += ws[O_RENT + base + i];
      c += ws[O_RCNT + base + i];
    }
    float S = blockReduce(s);
    float Q = blockReduce(q);
    float E = blockReduce(e);
    float C = blockReduce(c);
    if (tid == 0) {
      const double n = 33554432.0;
      double mean = (double)S / n;
      double var  = ((double)Q - (double)S * (double)S / n) / (n - 1.0);
      if (var < 0.0) var = 0.0;
      stats_s[m][0] = (float)mean;
      stats_s[m][1] = (float)sqrt(var);
      stats_s[m][2] = E / 16384.0f;          // mean entropy over H*N rows
      stats_s[m][3] = (float)((double)C / n);
    }
  }
  __syncthreads();

  if (tid == 0) {
    float lg[3];
    for (int m = 0; m < 3; ++m) {
      float acc = gb2[0];
      for (int j = 0; j < 64; ++j) {
        float hj = gb1[j];
        for (int i = 0; i < 4; ++i) hj = fmaf(stats_s[m][i], gw1[i * 64 + j], hj);
        hj = fmaxf(hj, 0.0f);
        acc = fmaf(hj, gw2[j], acc);
      }
      lg[m] = acc;
    }
    float mx = fmaxf(lg[0], fmaxf(lg[1], lg[2]));
    float ex0 = expf(lg[0] - mx), ex1 = expf(lg[1] - mx), ex2 = expf(lg[2] - mx);
    float isum = 1.0f / (ex0 + ex1 + ex2);
    alpha_s[0] = ex0 * isum; alpha_s[1] = ex1 * isum; alpha_s[2] = ex2 * isum;
    ws[O_ALPHA + 0] = alpha_s[0];
    ws[O_ALPHA + 1] = alpha_s[1];
    ws[O_ALPHA + 2] = alpha_s[2];
  }
  __syncthreads();

  const float a0 = alpha_s[0], a1 = alpha_s[1], a2 = alpha_s[2];
  for (int i = tid; i < 16384; i += 256) {
    float rs = a0 * ws[O_RSUM + i] + a1 * ws[O_RSUM + 16384 + i] + a2 * ws[O_RSUM + 32768 + i];
    ws[O_DINV + i] = 1.0f / fmaxf(rs, 1e-8f);
  }
}

// ================= pass 3: fused-attention GEMM via V_WMMA_F32_16X16X4_F32 ===
// grid = 8 heads * 64 row-tiles = 512 blocks, 256 threads = 8 waves.
// 32-row M-tile: each wave holds TWO 16x16 f32 accumulators sharing one
// B-fragment -> 2 WMMAs per B load, halves L2 traffic on H_split.
static constexpr int KC = 256;   // K-chunk staged in LDS
static constexpr int SP = 260;   // LDS row stride: banks 4r%64 distinct, 16B aligned
static constexpr int TM = 32;    // M-tile rows per block

__global__ void __launch_bounds__(256)
k_gemm(const float* __restrict__ a0p, const float* __restrict__ a1p,
       const float* __restrict__ a2p, float* __restrict__ ws) {
  __shared__ float lds[TM * SP];   // 33280 B
  const float* Hs = ws + O_HS;

  const float al0 = ws[O_ALPHA + 0];
  const float al1 = ws[O_ALPHA + 1];
  const float al2 = ws[O_ALPHA + 2];

  const int h  = blockIdx.x >> 6;        // 64 row-tiles per head
  const int it = blockIdx.x & 63;
  const int i0 = it * TM;
  const size_t rowbase = ((size_t)h * 2048 + i0) * 2048;

  const int tid  = threadIdx.x;
  const int lane = tid & 31;
  const int wave = tid >> 5;
  const int nrow = lane & 15;   // M for A-frag, N for B/C-frags
  const int half = lane >> 4;

  v8f acc0 = {0.f, 0.f, 0.f, 0.f, 0.f, 0.f, 0.f, 0.f};
  v8f acc1 = {0.f, 0.f, 0.f, 0.f, 0.f, 0.f, 0.f, 0.f};

  for (int k0 = 0; k0 < 2048; k0 += KC) {
    __syncthreads();  // protect LDS from previous chunk's readers
    // cooperative fused load: 32 rows x 256 cols as float4 (2048 f4 / 256 thr)
    for (int q = tid; q < 2048; q += 256) {
      const int r  = q >> 6;             // 64 float4 per row
      const int c4 = q & 63;
      const size_t off = rowbase + (size_t)r * 2048 + (size_t)k0 + (size_t)c4 * 4;
      v4f x0 = __builtin_nontemporal_load((const v4f*)(a0p + off));
      v4f x1 = __builtin_nontemporal_load((const v4f*)(a1p + off));
      v4f x2 = __builtin_nontemporal_load((const v4f*)(a2p + off));
      v4f f = al0 * x0 + al1 * x1 + al2 * x2;
      *(v4f*)&lds[r * SP + c4 * 4] = f;
    }
    __syncthreads();

    const float* bcol = Hs + (size_t)k0 * 128 + wave * 16 + nrow;
    for (int k = 0; k < KC; k += 4) {
      const int kk = k + 2 * half;
      // B-frag: lane holds B[kk][N], B[kk+1][N] -- shared by both M-halves
      v2f b = {bcol[(size_t)kk * 128], bcol[(size_t)(kk + 1) * 128]};
      // A-frags: rows nrow (acc0) and nrow+16 (acc1)
      v2f aa0 = *(const v2f*)&lds[nrow * SP + kk];
      v2f aa1 = *(const v2f*)&lds[(nrow + 16) * SP + kk];
      acc0 = __builtin_amdgcn_wmma_f32_16x16x4_f32(
          false, aa0, false, b, (short)0, acc0, false, false);
      acc1 = __builtin_amdgcn_wmma_f32_16x16x4_f32(
          false, aa1, false, b, (short)0, acc1, false, false);
    }
  }

  // writeout: C layout — lane,vgpr v: M = v + 8*half, N = nrow
  const int d = wave * 16 + nrow;
  for (int v = 0; v < 8; ++v) {
    const int M0  = v + 8 * half;          // rows i0 .. i0+15
    const int gi0 = h * 2048 + i0 + M0;
    ws[O_PART + (size_t)gi0 * 128 + d] = acc0[v] * ws[O_DINV + gi0] * 0.125f;
    const int gi1 = gi0 + 16;              // rows i0+16 .. i0+31
    ws[O_PART + (size_t)gi1 * 128 + d] = acc1[v] * ws[O_DINV + gi1] * 0.125f;
  }
}

// ================= pass 4: head-sum + refine MLP + residual + LayerNorm ======
// grid = 2048 blocks (one per output row), 128 threads (one per channel).
__global__ void __launch_bounds__(128)
k_final(const float* __restrict__ refine_w, const float* __restrict__ refine_b,
        const float* __restrict__ ln_g, const float* __restrict__ ln_b,
        const float* __restrict__ ws, float* __restrict__ out) {
  __shared__ float arow[128];
  __shared__ float red[128];
  const int i = blockIdx.x;
  const int d = threadIdx.x;

  float agg = 0.f;
  for (int h = 0; h < 8; ++h)
    agg += ws[O_PART + ((size_t)(h * 2048 + i)) * 128 + d];
  arow[d] = agg;
  __syncthreads();

  float pre = refine_b[d];
  const float* wr = refine_w + (size_t)d * 128;   // (agg @ W^T)[d] = <agg, W[d,:]>
  for (int k = 0; k < 128; ++k) pre = fmaf(arow[k], wr[k], pre);

  const float o = fmaxf(pre, 0.0f) + ws[O_HS + (size_t)i * 128 + d];

  red[d] = o; __syncthreads();
  for (int off = 64; off > 0; off >>= 1) {
    if (d < off) red[d] += red[d + off];
    __syncthreads();
  }
  const float mu = red[0] * (1.0f / 128.0f);
  __syncthreads();

  const float c = o - mu;
  red[d] = c * c; __syncthreads();
  for (int off = 64; off > 0; off >>= 1) {
    if (d < off) red[d] += red[d + off];
    __syncthreads();
  }
  const float var = red[0] * (1.0f / 128.0f);

  out[(size_t)i * 128 + d] = c * rsqrtf(var + 1e-5f) * ln_g[d] + ln_b[d];
}

// =============================================================================
extern "C" void kernel_launch(void* const* d_in, const int* in_sizes, int n_in,
                              void* d_out, int out_size, void* d_ws, size_t ws_size,
                              hipStream_t stream) {
  // setup_inputs() dict order (interleaved embeds/attn):
  const float* e0  = (const float*)d_in[0];
  const float* a0  = (const float*)d_in[1];
  const float* e1  = (const float*)d_in[2];
  const float* a1  = (const float*)d_in[3];
  const float* e2  = (const float*)d_in[4];
  const float* a2  = (const float*)d_in[5];
  const float* gw1 = (const float*)d_in[6];
  const float* gb1 = (const float*)d_in[7];
  const float* gw2 = (const float*)d_in[8];
  const float* gb2 = (const float*)d_in[9];
  const float* rw  = (const float*)d_in[10];
  const float* rb  = (const float*)d_in[11];
  const float* lg  = (const float*)d_in[12];
  const float* lb  = (const float*)d_in[13];
  float* ws  = (float*)d_ws;
  float* out = (float*)d_out;

  k_stats <<<49152, 256, 0, stream>>>(a0, a1, a2, ws);
  k_hsplit<<<  256, 256, 0, stream>>>(e0, e1, e2, ws);
  k_gate  <<<    1, 256, 0, stream>>>(gw1, gb1, gw2, gb2, ws);
  k_gemm  <<<  512, 256, 0, stream>>>(a0, a1, a2, ws);
  k_final <<< 2048, 128, 0, stream>>>(rw, rb, lg, lb, ws, out);
}